// SPRGNN_88648124990914
// MI455X (gfx1250) — compile-verified
//
#include <hip/hip_runtime.h>

#define NNODES 100000
#define NGRAPHS 1024
#define EMB 32
#define HID 64
#define NCLASSES 10

typedef __attribute__((ext_vector_type(2))) float v2f;
typedef __attribute__((ext_vector_type(8))) float v8f;

// ---------------------------------------------------------------- utilities
__global__ void zero_f32(float* __restrict__ p, long n) {
  long i = (long)blockIdx.x * blockDim.x + threadIdx.x;
  if (i < n) p[i] = 0.0f;
}

// deg[n] = #edges with dst == n   (u32 atomics into float-aliased buffer)
__global__ void deg_kernel(const int* __restrict__ dst, unsigned int* __restrict__ deg, int E) {
  int e = blockIdx.x * blockDim.x + threadIdx.x;
  if (e < E) atomicAdd(&deg[dst[e]], 1u);
}

// in-place: dis[n] = rsqrt(1 + deg[n])
__global__ void dis_kernel(float* __restrict__ disbuf, int n) {
  int i = blockIdx.x * blockDim.x + threadIdx.x;
  if (i < n) {
    unsigned int d = ((const unsigned int*)disbuf)[i];
    disbuf[i] = rsqrtf(1.0f + (float)d);
  }
}

// x0[n,c] = shape_table[sid][c]*(sid!=0) + color_table[cid][c]*(cid!=0)
__global__ void embed_kernel(const int* __restrict__ sid, const int* __restrict__ cid,
                             const float* __restrict__ st, const float* __restrict__ ct,
                             float* __restrict__ x0, int n) {
  long i = (long)blockIdx.x * blockDim.x + threadIdx.x;
  if (i >= (long)n * EMB) return;
  int node = (int)(i >> 5);
  int c = (int)(i & 31);
  int s = sid[node], cc = cid[node];
  float v = 0.0f;
  if (s != 0) v += st[s * EMB + c];
  if (cc != 0) v += ct[cc * EMB + c];
  x0[i] = v;
}

// ------------------------------------------------- WMMA fp32 GEMM  H = X @ W
// X: [NNODES, K] row-major, W: [K, HID] row-major, H: [NNODES, HID].
// grid.x = NNODES/16 (exact), block = 128 (4 waves), wave w -> column tile w.
// V_WMMA_F32_16X16X4_F32 fragments (wave32):
//   A (16x4): lane L: M=L&15, elems j=0,1 -> K = 2*(L>>4)+j
//   B (4x16): lane L: N=L&15, elems j=0,1 -> K = 2*(L>>4)+j
//   C/D v8f : lane L: N=L&15, elem v -> M = v + 8*(L>>4)
__global__ void gemm_wmma_f32(const float* __restrict__ X, const float* __restrict__ W,
                              float* __restrict__ H, int K) {
  const int lane  = threadIdx.x & 31;
  const int wave  = threadIdx.x >> 5;      // 0..3 => column tile
  const int l16   = lane & 15;
  const int khalf = lane >> 4;             // 0 or 1
  const long row  = (long)blockIdx.x * 16 + l16;
  const int  col  = wave * 16 + l16;

  v8f c = {};
  const float* ap = X + row * K + 2 * khalf;
  for (int k0 = 0; k0 < K; k0 += 4) {
    v2f a, b;
    a.x = ap[k0];
    a.y = ap[k0 + 1];
    b.x = W[(k0 + 2 * khalf + 0) * HID + col];
    b.y = W[(k0 + 2 * khalf + 1) * HID + col];
    c = __builtin_amdgcn_wmma_f32_16x16x4_f32(false, a, false, b, (short)0, c, false, false);
  }
  const long rbase = (long)blockIdx.x * 16 + 8 * khalf;
#pragma unroll
  for (int v = 0; v < 8; ++v)
    H[(rbase + v) * HID + col] = c[v];
}

// ------------------------------------------- edge scatter: agg[dst] += h[src]*norm
// 16 lanes per edge, 4 floats each (float4 gather, 4 f32 atomics).
__global__ void edge_agg_kernel(const int* __restrict__ src, const int* __restrict__ dst,
                                const float* __restrict__ h, const float* __restrict__ dis,
                                float* __restrict__ agg, int E) {
  long t = (long)blockIdx.x * blockDim.x + threadIdx.x;
  long e = t >> 4;
  if (e >= E) return;
  int c4 = (int)(t & 15) << 2;
  int s = src[e];
  int d = dst[e];
  // prefetch a future source row to keep L2/L0 ahead of the gather stream
  if (e + 512 < E) __builtin_prefetch(h + (long)src[e + 512] * HID + c4, 0, 1);
  float norm = dis[s] * dis[d];
  const float4 hv = *(const float4*)(h + (long)s * HID + c4);
  float* ap = agg + (long)d * HID + c4;
  atomicAdd(ap + 0, hv.x * norm);
  atomicAdd(ap + 1, hv.y * norm);
  atomicAdd(ap + 2, hv.z * norm);
  atomicAdd(ap + 3, hv.w * norm);
}

// x = relu(agg + h*dis^2 + b)   (in place into agg)
__global__ void finalize_kernel(float* __restrict__ agg, const float* __restrict__ h,
                                const float* __restrict__ dis, const float* __restrict__ b,
                                int n) {
  long i = (long)blockIdx.x * blockDim.x + threadIdx.x;
  if (i >= (long)n * HID) return;
  int node = (int)(i >> 6);
  int c = (int)(i & 63);
  float di = dis[node];
  float v = agg[i] + h[i] * di * di + b[c];
  agg[i] = v > 0.0f ? v : 0.0f;
}

// pooled[batch[n]] += x[n] ; cnt[batch[n]] += 1
__global__ void pool_kernel(const float* __restrict__ x, const int* __restrict__ batch,
                            float* __restrict__ pooled, float* __restrict__ cnt, int n) {
  long i = (long)blockIdx.x * blockDim.x + threadIdx.x;
  if (i >= (long)n * HID) return;
  int node = (int)(i >> 6);
  int c = (int)(i & 63);
  int g = batch[node];
  atomicAdd(pooled + (long)g * HID + c, x[i]);
  if (c == 0) atomicAdd(cnt + g, 1.0f);
}

// out[g,k] = (sum_h pooled[g,h]*Wl[h,k]) / max(cnt[g],1) + bl[k]
__global__ void out_kernel(const float* __restrict__ pooled, const float* __restrict__ cnt,
                           const float* __restrict__ Wl, const float* __restrict__ bl,
                           float* __restrict__ out) {
  int i = blockIdx.x * blockDim.x + threadIdx.x;
  if (i >= NGRAPHS * NCLASSES) return;
  int g = i / NCLASSES;
  int k = i % NCLASSES;
  float s = 0.0f;
#pragma unroll 8
  for (int hh = 0; hh < HID; ++hh) s += pooled[g * HID + hh] * Wl[hh * NCLASSES + k];
  float c = cnt[g];
  c = c > 1.0f ? c : 1.0f;
  out[i] = s / c + bl[k];
}

// ---------------------------------------------------------------- launcher
extern "C" void kernel_launch(void* const* d_in, const int* in_sizes, int n_in,
                              void* d_out, int out_size, void* d_ws, size_t ws_size,
                              hipStream_t stream) {
  const int* shape_id = (const int*)d_in[0];
  const int* color_id = (const int*)d_in[1];
  const int* edge_index = (const int*)d_in[2];
  const int* batch = (const int*)d_in[3];
  const float* shape_table = (const float*)d_in[4];
  const float* color_table = (const float*)d_in[5];
  const float* W1 = (const float*)d_in[6];
  const float* b1 = (const float*)d_in[7];
  const float* W2 = (const float*)d_in[8];
  const float* b2 = (const float*)d_in[9];
  const float* Wl = (const float*)d_in[10];
  const float* bl = (const float*)d_in[11];
  float* out = (float*)d_out;

  const int N = NNODES;
  const int E = in_sizes[2] / 2;
  const int* src = edge_index;
  const int* dst = edge_index + E;

  // workspace layout (floats)
  float* dis    = (float*)d_ws;                 // N        (aliased as u32 degree first)
  float* x0     = dis + N;                      // N*EMB
  float* bufH   = x0 + (long)N * EMB;           // N*HID
  float* bufA   = bufH + (long)N * HID;         // N*HID
  float* pooled = bufA + (long)N * HID;         // NGRAPHS*HID
  float* cnt    = pooled + (long)NGRAPHS * HID; // NGRAPHS
  (void)n_in; (void)out_size; (void)ws_size;

  const int B = 256;
  auto blocks = [](long n, int b) { return (unsigned)((n + b - 1) / b); };

  // 1) symmetric-norm coefficients dis = rsqrt(1+deg)
  zero_f32<<<blocks(N, B), B, 0, stream>>>(dis, N);
  deg_kernel<<<blocks(E, B), B, 0, stream>>>(dst, (unsigned int*)dis, E);
  dis_kernel<<<blocks(N, B), B, 0, stream>>>(dis, N);

  // 2) embedding lookup (padding_idx = 0)
  embed_kernel<<<blocks((long)N * EMB, B), B, 0, stream>>>(shape_id, color_id,
                                                           shape_table, color_table, x0, N);

  // 3) layer 1: h = x0 @ W1 (WMMA), aggregate, fused self-loop+bias+relu
  gemm_wmma_f32<<<N / 16, 128, 0, stream>>>(x0, W1, bufH, EMB);
  zero_f32<<<blocks((long)N * HID, B), B, 0, stream>>>(bufA, (long)N * HID);
  edge_agg_kernel<<<blocks((long)E * 16, B), B, 0, stream>>>(src, dst, bufH, dis, bufA, E);
  finalize_kernel<<<blocks((long)N * HID, B), B, 0, stream>>>(bufA, bufH, dis, b1, N);

  // 4) layer 2: h = x1 @ W2 (WMMA), aggregate, fused self-loop+bias+relu
  gemm_wmma_f32<<<N / 16, 128, 0, stream>>>(bufA, W2, bufH, HID);
  zero_f32<<<blocks((long)N * HID, B), B, 0, stream>>>(bufA, (long)N * HID);
  edge_agg_kernel<<<blocks((long)E * 16, B), B, 0, stream>>>(src, dst, bufH, dis, bufA, E);
  finalize_kernel<<<blocks((long)N * HID, B), B, 0, stream>>>(bufA, bufH, dis, b2, N);

  // 5) global mean pool + classifier head
  zero_f32<<<blocks((long)NGRAPHS * HID + NGRAPHS, B), B, 0, stream>>>(pooled,
                                                                      (long)NGRAPHS * HID + NGRAPHS);
  pool_kernel<<<blocks((long)N * HID, B), B, 0, stream>>>(bufA, batch, pooled, cnt, N);
  out_kernel<<<blocks((long)NGRAPHS * NCLASSES, B), B, 0, stream>>>(pooled, cnt, Wl, bl, out);
}